// LitePTDetectionHead_47699906789824
// MI455X (gfx1250) — compile-verified
//
#include <hip/hip_runtime.h>
#include <hip/hip_bf16.h>

#define Bc     8
#define Pc     125000
#define NCc    6
#define NCAND  400
#define MAXB   100
#define GCAP   4096
#define NEGV   (-1.0e9f)

typedef __attribute__((ext_vector_type(2))) float v2f;
typedef __attribute__((ext_vector_type(8))) float v8f;

// ---------------------------------------------------------------- utilities
__global__ void zero_kernel(float* p, long n) {
    long i = (long)blockIdx.x * blockDim.x + threadIdx.x;
    if (i < n) p[i] = 0.0f;
}

// ---------------------------------------------------------------- top-k: histogram / threshold / gather / rank-select
__global__ void hist_kernel(const float* __restrict__ scores, int* __restrict__ hist) {
    long i = (long)blockIdx.x * blockDim.x + threadIdx.x;
    if (i >= (long)Bc * Pc) return;
    int b = (int)(i / Pc);
    float s = scores[i];
    if (s > 0.01f) {
        int bin = (int)(s * 1024.0f);
        bin = bin < 0 ? 0 : (bin > 1023 ? 1023 : bin);
        atomicAdd(&hist[b * 1024 + bin], 1);
    }
}

__global__ void thresh_kernel(const int* __restrict__ hist, int* __restrict__ bth) {
    int b = threadIdx.x;
    if (b >= Bc) return;
    int cum = 0, thr = 0;
    for (int t = 1023; t >= 0; t--) {
        cum += hist[b * 1024 + t];
        if (cum >= NCAND) { thr = t; break; }
    }
    bth[b] = thr;
}

__global__ void gather_kernel(const float* __restrict__ scores, const int* __restrict__ bth,
                              int* __restrict__ gcnt, int* __restrict__ gidx) {
    long i = (long)blockIdx.x * blockDim.x + threadIdx.x;
    if (i >= (long)Bc * Pc) return;
    int b = (int)(i / Pc);
    float s = scores[i];
    if (s > 0.01f) {
        int bin = (int)(s * 1024.0f);
        bin = bin < 0 ? 0 : (bin > 1023 ? 1023 : bin);
        if (bin >= bth[b]) {
            int pos = atomicAdd(&gcnt[b], 1);
            if (pos < GCAP) gidx[b * GCAP + pos] = (int)(i - (long)b * Pc);
        }
    }
}

__global__ void __launch_bounds__(1024)
rank_select_kernel(const float* __restrict__ scores, const int* __restrict__ gcnt,
                   const int* __restrict__ gidx, int* __restrict__ top_idx,
                   float* __restrict__ top_score, int* __restrict__ valid) {
    __shared__ float ls[GCAP];
    __shared__ int   li[GCAP];
    int b = blockIdx.x;
    int M = gcnt[b]; if (M > GCAP) M = GCAP;
    for (int k = threadIdx.x; k < M; k += blockDim.x) {
        int idx = gidx[b * GCAP + k];
        li[k] = idx;
        ls[k] = scores[(long)b * Pc + idx];
    }
    for (int k = threadIdx.x; k < NCAND; k += blockDim.x) {
        top_idx[b * NCAND + k] = 0;
        top_score[b * NCAND + k] = NEGV;
        valid[b * NCAND + k] = 0;
    }
    __syncthreads();
    for (int k = threadIdx.x; k < M; k += blockDim.x) {
        float si = ls[k]; int ii = li[k];
        int rank = 0;
        for (int j = 0; j < M; j++) {
            float sj = ls[j];
            rank += ((sj > si) || (sj == si && li[j] < ii)) ? 1 : 0;
        }
        if (rank < NCAND) {
            top_idx[b * NCAND + rank] = ii;
            top_score[b * NCAND + rank] = si;
            valid[b * NCAND + rank] = 1;
        }
    }
}

// ---------------------------------------------------------------- candidate prep (boxes, labels, WMMA coeffs, corners)
__global__ void prep_cand_kernel(const float* __restrict__ box_preds, const float* __restrict__ cls_preds,
                                 const int* __restrict__ top_idx, const int* __restrict__ valid,
                                 float* __restrict__ cand_box, float* __restrict__ cand_cls,
                                 int* __restrict__ cand_label, float* __restrict__ cinfo,
                                 float* __restrict__ corners, float* __restrict__ areas) {
    int t = blockIdx.x * blockDim.x + threadIdx.x;
    if (t >= Bc * NCAND) return;
    int b = t / NCAND;
    int idx = top_idx[t];
    const float* bp = box_preds + ((long)b * Pc + idx) * 7;
    float x = bp[0], y = bp[1], z = bp[2], dx = bp[3], dy = bp[4], dz = bp[5], h = bp[6];
    float* cb = cand_box + (long)t * 7;
    cb[0] = x; cb[1] = y; cb[2] = z; cb[3] = dx; cb[4] = dy; cb[5] = dz; cb[6] = h;

    const float* cl = cls_preds + ((long)b * Pc + idx) * NCc;
    float* cc = cand_cls + (long)t * NCc;
    float bv = cl[0]; int bl = 0; cc[0] = cl[0];
    for (int k = 1; k < NCc; k++) { cc[k] = cl[k]; if (cl[k] > bv) { bv = cl[k]; bl = k; } }
    cand_label[t] = bl;

    float c = cosf(h), s = sinf(h);
    float* ci = cinfo + (long)t * 12;
    ci[0] = c; ci[1] = s;
    ci[2] = -(c * x + s * y);       // dlx:  lx = c*px + s*py + dlx
    ci[3] = s * x - c * y;          // dly:  ly = -s*px + c*py + dly
    ci[4] = z;                      // bz:   dz = pz - bz
    ci[5] = 0.5f * dx; ci[6] = 0.5f * dy; ci[7] = 0.5f * dz;
    ci[8] = valid[t] ? 1.0f : 0.0f;
    ci[9] = ci[10] = ci[11] = 0.0f;

    const float sgx[4] = {1.f, -1.f, -1.f, 1.f};
    const float sgy[4] = {1.f, 1.f, -1.f, -1.f};
    float* co = corners + (long)t * 8;
    for (int k = 0; k < 4; k++) {
        float lx = 0.5f * dx * sgx[k], ly = 0.5f * dy * sgy[k];
        co[2 * k]     = x + c * lx - s * ly;
        co[2 * k + 1] = y + s * lx + c * ly;
    }
    areas[t] = dx * dy;
}

// ---------------------------------------------------------------- WMMA point-in-box assignment
// One wave handles a 16-point tile: A(16x4) = [px,py,pz,1], B(4x16) = per-candidate
// affine coefficients, D(16x16) = lx / ly / dz via V_WMMA_F32_16X16X4_F32.
// The "first containing candidate" (min index) per matrix row is a priority encode
// over the 16 lanes holding that row: one wave32 ballot gives candidates for row r
// in bits[15:0] and row r+8 in bits[31:16] simultaneously.
__global__ void __launch_bounds__(256)
assign_kernel(const float* __restrict__ coords, const float* __restrict__ cinfo,
              int* __restrict__ assigned) {
    const int numTiles = (Pc + 15) / 16;
    int wave = threadIdx.x >> 5;
    int lane = threadIdx.x & 31;
    int tile = blockIdx.x * (blockDim.x >> 5) + wave;
    int b = blockIdx.y;
    if (tile >= numTiles) return;           // wave-uniform exit (EXEC stays all-ones for WMMA)

    int half = lane >> 4;                   // 0: lanes 0-15 (K=0,1) ; 1: lanes 16-31 (K=2,3)
    int l15 = lane & 15;
    int row = tile * 16 + l15;              // point row M for the A matrix
    int prow = row < Pc ? row : Pc - 1;     // clamp; out-of-range rows masked at write
    const float* cp = coords + ((long)b * Pc + prow) * 3;

    v2f A;
    if (half == 0) { A.x = cp[0]; A.y = cp[1]; }   // K0=px, K1=py
    else           { A.x = cp[2]; A.y = 1.0f; }    // K2=pz, K3=1

    int best[8];
#pragma unroll
    for (int r = 0; r < 8; r++) best[r] = 0x7fffffff;

    const float* cib = cinfo + (long)b * NCAND * 12;
    for (int t = 0; t < NCAND / 16; t++) {
        int n = t * 16 + l15;               // candidate column N held by this lane
        const float* ci = cib + (long)n * 12;
        float c = ci[0], s = ci[1], dlx = ci[2], dly = ci[3], bz = ci[4];
        float hx = ci[5], hy = ci[6], hz = ci[7], vf = ci[8];

        v2f Blx, Bly, Bdz;
        if (half == 0) { Blx.x = c;   Blx.y = s;   Bly.x = -s;  Bly.y = c;   Bdz.x = 0.f; Bdz.y = 0.f; }
        else           { Blx.x = 0.f; Blx.y = dlx; Bly.x = 0.f; Bly.y = dly; Bdz.x = 1.f; Bdz.y = -bz; }

        v8f z = {};
        v8f LX = __builtin_amdgcn_wmma_f32_16x16x4_f32(false, A, false, Blx, (short)0, z, false, false);
        v8f LY = __builtin_amdgcn_wmma_f32_16x16x4_f32(false, A, false, Bly, (short)0, z, false, false);
        v8f DZ = __builtin_amdgcn_wmma_f32_16x16x4_f32(false, A, false, Bdz, (short)0, z, false, false);

        bool cv = vf > 0.5f;
#pragma unroll
        for (int r = 0; r < 8; r++) {
            bool inside = cv && (fabsf(LX[r]) <= hx) && (fabsf(LY[r]) <= hy) && (fabsf(DZ[r]) <= hz);
            unsigned mask = (unsigned)__ballot(inside);          // wave32 ballot
            unsigned hm = half ? (mask >> 16) : (mask & 0xffffu);
            int j = hm ? (t * 16 + (__ffs(hm) - 1)) : 0x7fffffff;
            best[r] = j < best[r] ? j : best[r];
        }
    }
    // Row r lives in lanes 0-15 (point tile*16+r) and r+8 in lanes 16-31.
    // Two lanes (r and 16+r) emit each row pair; static register indexing.
    long basep = (long)b * Pc + tile * 16 + half * 8;
#pragma unroll
    for (int r = 0; r < 8; r++) {
        if (l15 == r) {
            int p = tile * 16 + r + half * 8;
            if (p < Pc)
                assigned[basep + r] = (best[r] == 0x7fffffff) ? -1 : best[r];
        }
    }
}

// ---------------------------------------------------------------- seg consistency stats
__global__ void seg_kernel(const float* __restrict__ seg_logits, const int* __restrict__ assigned,
                           const int* __restrict__ cand_label, float* __restrict__ total,
                           float* __restrict__ cons) {
    long i = (long)blockIdx.x * blockDim.x + threadIdx.x;
    if (i >= (long)Bc * Pc) return;
    int j = assigned[i];
    if (j < 0) return;
    int b = (int)(i / Pc);
    const float* sl = seg_logits + i * NCc;
    float bv = sl[0]; int lab = 0;
    for (int k = 1; k < NCc; k++) if (sl[k] > bv) { bv = sl[k]; lab = k; }
    atomicAdd(&total[b * NCAND + j], 1.0f);
    if (lab == cand_label[b * NCAND + j]) atomicAdd(&cons[b * NCAND + j], 1.0f);
}

__global__ void refine_kernel(const float* __restrict__ top_score, const float* __restrict__ total,
                              const float* __restrict__ cons, float* __restrict__ ref) {
    int t = blockIdx.x * blockDim.x + threadIdx.x;
    if (t >= Bc * NCAND) return;
    float tt = total[t], cc = cons[t];
    float occ = cc / fmaxf(tt, 1.0f);
    float mult = (0.5f + 0.5f * occ) * (1.0f + log1pf(cc) * 0.2f);
    ref[t] = top_score[t] * mult;
}

// ---------------------------------------------------------------- rotated-box IoU (Sutherland–Hodgman, mirrors reference)
__device__ inline void clip_edge(float* px, float* py, int& n,
                                 float p1x, float p1y, float p2x, float p2y) {
    float ex = p2x - p1x, ey = p2y - p1y;
    float ox[8], oy[8];
    for (int k = 0; k < 8; k++) { ox[k] = 0.f; oy[k] = 0.f; }
    int m = 0;
    for (int i = 0; i < 8; i++) {
        int nn = n > 1 ? n : 1;
        float cx = px[i], cy = py[i];
        int ni = (i + 1) % nn;
        float nx = px[ni], ny = py[ni];
        float fc = ex * (cy - p1y) - ey * (cx - p1x);
        float fn = ex * (ny - p1y) - ey * (nx - p1x);
        bool active = i < n;
        bool inc = fc >= 0.0f, inn = fn >= 0.0f;
        bool add1 = active && inc;
        if (add1 && m < 8) { ox[m] = cx; oy[m] = cy; }
        m += add1 ? 1 : 0;
        float den = fc - fn;
        if (fabsf(den) < 1e-12f) den = 1e-12f;
        float ipx = cx + fc / den * (nx - cx);
        float ipy = cy + fc / den * (ny - cy);
        bool add2 = active && (inc != inn);
        if (add2 && m < 8) { ox[m] = ipx; oy[m] = ipy; }
        m += add2 ? 1 : 0;
    }
    for (int k = 0; k < 8; k++) { px[k] = ox[k]; py[k] = oy[k]; }
    n = m;
}

__global__ void iou_kernel(const float* __restrict__ corners, const float* __restrict__ areas,
                           float* __restrict__ iou) {
    long t = (long)blockIdx.x * blockDim.x + threadIdx.x;
    if (t >= (long)Bc * NCAND * NCAND) return;
    int b = (int)(t / (NCAND * NCAND));
    int rem = (int)(t % (NCAND * NCAND));
    int i = rem / NCAND, jj = rem % NCAND;
    const float* ca = corners + ((long)b * NCAND + i) * 8;
    const float* cb = corners + ((long)b * NCAND + jj) * 8;
    float aa = areas[b * NCAND + i], ab = areas[b * NCAND + jj];
    float px[8], py[8];
    for (int k = 0; k < 8; k++) { px[k] = 0.f; py[k] = 0.f; }
    for (int k = 0; k < 4; k++) { px[k] = ca[2 * k]; py[k] = ca[2 * k + 1]; }
    int n = 4;
    for (int e = 0; e < 4; e++) {
        int e2 = (e + 1) & 3;
        clip_edge(px, py, n, cb[2 * e], cb[2 * e + 1], cb[2 * e2], cb[2 * e2 + 1]);
    }
    int nn = n > 1 ? n : 1;
    float sum = 0.0f;
    for (int k = 0; k < 8; k++) {
        int nk = (k + 1) % nn;
        if (k < n) sum += px[k] * py[nk] - py[k] * px[nk];
    }
    float inter = 0.5f * fabsf(sum);
    iou[t] = inter / fmaxf(aa + ab - inter, 1e-6f);
}

// ---------------------------------------------------------------- stable argsort(-ref)
__global__ void order_kernel(const float* __restrict__ ref, int* __restrict__ order) {
    __shared__ float r[NCAND];
    int b = blockIdx.x;
    for (int i = threadIdx.x; i < NCAND; i += blockDim.x) r[i] = ref[b * NCAND + i];
    __syncthreads();
    for (int i = threadIdx.x; i < NCAND; i += blockDim.x) {
        float ri = r[i];
        int rank = 0;
        for (int j = 0; j < NCAND; j++) {
            float rj = r[j];
            rank += ((rj > ri) || (rj == ri && j < i)) ? 1 : 0;
        }
        order[b * NCAND + rank] = i;
    }
}

// ---------------------------------------------------------------- NMS + output packing
__global__ void nms_out_kernel(const int* __restrict__ order, const int* __restrict__ valid,
                               const int* __restrict__ cand_label, const float* __restrict__ iou,
                               const float* __restrict__ ref, const float* __restrict__ cand_box,
                               const float* __restrict__ cand_cls, float* __restrict__ out) {
    __shared__ int ord[NCAND], slab[NCAND], sval[NCAND], sup[NCAND], pos[NCAND];
    int b = blockIdx.x, tid = threadIdx.x, bd = blockDim.x;
    for (int i = tid; i < NCAND; i += bd) {
        int o = order[b * NCAND + i];
        ord[i] = o;
        slab[i] = cand_label[b * NCAND + o];
        sval[i] = valid[b * NCAND + o];
        sup[i] = 0;
    }
    __syncthreads();
    const float* ioub = iou + (long)b * NCAND * NCAND;
    for (int i = 0; i < NCAND; i++) {
        bool keep = sval[i] && !sup[i];
        if (keep) {
            int oi = ord[i], li = slab[i];
            for (int j = i + 1 + tid; j < NCAND; j += bd) {
                if (slab[j] == li && ioub[oi * NCAND + ord[j]] > 0.7f) sup[j] = 1;
            }
        }
        __syncthreads();
    }
    if (tid == 0) {
        int cum = 0;
        for (int i = 0; i < NCAND; i++) {
            bool kept = sval[i] && !sup[i];
            pos[i] = kept ? cum : -1;
            cum += kept ? 1 : 0;
        }
    }
    __syncthreads();
    float* ob = out;                         // [B,100,7]
    float* os = out + Bc * MAXB * 7;         // [B,100]
    float* oc = os + Bc * MAXB;              // [B,100,6]
    float* ov = oc + Bc * MAXB * NCc;        // [B,100]
    for (int i = tid; i < NCAND; i += bd) {
        int p = pos[i];
        if (p >= 0 && p < MAXB) {
            int o = ord[i];
            const float* cbx = cand_box + ((long)b * NCAND + o) * 7;
            for (int k = 0; k < 7; k++) ob[((long)b * MAXB + p) * 7 + k] = cbx[k];
            os[b * MAXB + p] = ref[b * NCAND + o];
            const float* ccl = cand_cls + ((long)b * NCAND + o) * NCc;
            for (int k = 0; k < NCc; k++) oc[((long)b * MAXB + p) * NCc + k] = ccl[k];
            ov[b * MAXB + p] = 1.0f;
        }
    }
}

// ---------------------------------------------------------------- launcher
extern "C" void kernel_launch(void* const* d_in, const int* in_sizes, int n_in,
                              void* d_out, int out_size, void* d_ws, size_t ws_size,
                              hipStream_t stream) {
    const float* box_preds  = (const float*)d_in[0];
    const float* cls_scores = (const float*)d_in[1];
    const float* cls_preds  = (const float*)d_in[2];
    const float* seg_logits = (const float*)d_in[3];
    const float* coords     = (const float*)d_in[4];
    float* out = (float*)d_out;

    char* w = (char*)d_ws;
    size_t off = 0;
    auto alloc = [&](size_t nbytes) -> char* {
        char* p = w + off;
        off += (nbytes + 255) & ~(size_t)255;
        return p;
    };
    int*   hist       = (int*)alloc((size_t)Bc * 1024 * 4);
    int*   bth        = (int*)alloc(Bc * 4);
    int*   gcnt       = (int*)alloc(Bc * 4);
    int*   gidx       = (int*)alloc((size_t)Bc * GCAP * 4);
    int*   top_idx    = (int*)alloc((size_t)Bc * NCAND * 4);
    int*   valid      = (int*)alloc((size_t)Bc * NCAND * 4);
    int*   cand_label = (int*)alloc((size_t)Bc * NCAND * 4);
    int*   order      = (int*)alloc((size_t)Bc * NCAND * 4);
    int*   assigned   = (int*)alloc((size_t)Bc * Pc * 4);
    float* top_score  = (float*)alloc((size_t)Bc * NCAND * 4);
    float* cand_box   = (float*)alloc((size_t)Bc * NCAND * 7 * 4);
    float* cand_cls   = (float*)alloc((size_t)Bc * NCAND * NCc * 4);
    float* cinfo      = (float*)alloc((size_t)Bc * NCAND * 12 * 4);
    float* corners    = (float*)alloc((size_t)Bc * NCAND * 8 * 4);
    float* areas      = (float*)alloc((size_t)Bc * NCAND * 4);
    float* total      = (float*)alloc((size_t)Bc * NCAND * 4);
    float* cons       = (float*)alloc((size_t)Bc * NCAND * 4);
    float* refsc      = (float*)alloc((size_t)Bc * NCAND * 4);
    float* iou        = (float*)alloc((size_t)Bc * NCAND * NCAND * 4);

    const long NP = (long)Bc * Pc;
    const int TB = 256;

    // zero: outputs, histograms, counters, stats
    zero_kernel<<<(out_size + TB - 1) / TB, TB, 0, stream>>>(out, out_size);
    zero_kernel<<<(Bc * 1024 + TB - 1) / TB, TB, 0, stream>>>((float*)hist, Bc * 1024);
    zero_kernel<<<1, TB, 0, stream>>>((float*)gcnt, Bc);
    zero_kernel<<<(Bc * NCAND * 2 + TB - 1) / TB, TB, 0, stream>>>(total, Bc * NCAND * 2); // total+cons contiguous

    hist_kernel<<<(NP + TB - 1) / TB, TB, 0, stream>>>(cls_scores, hist);
    thresh_kernel<<<1, 32, 0, stream>>>(hist, bth);
    gather_kernel<<<(NP + TB - 1) / TB, TB, 0, stream>>>(cls_scores, bth, gcnt, gidx);
    rank_select_kernel<<<Bc, 1024, 0, stream>>>(cls_scores, gcnt, gidx, top_idx, top_score, valid);
    prep_cand_kernel<<<(Bc * NCAND + TB - 1) / TB, TB, 0, stream>>>(
        box_preds, cls_preds, top_idx, valid, cand_box, cand_cls, cand_label, cinfo, corners, areas);

    const int numTiles = (Pc + 15) / 16;
    dim3 agrid((numTiles + 7) / 8, Bc);
    assign_kernel<<<agrid, 256, 0, stream>>>(coords, cinfo, assigned);

    seg_kernel<<<(NP + TB - 1) / TB, TB, 0, stream>>>(seg_logits, assigned, cand_label, total, cons);
    refine_kernel<<<(Bc * NCAND + TB - 1) / TB, TB, 0, stream>>>(top_score, total, cons, refsc);

    long npairs = (long)Bc * NCAND * NCAND;
    iou_kernel<<<(unsigned)((npairs + TB - 1) / TB), TB, 0, stream>>>(corners, areas, iou);

    order_kernel<<<Bc, 512, 0, stream>>>(refsc, order);
    nms_out_kernel<<<Bc, 512, 0, stream>>>(order, valid, cand_label, iou, refsc, cand_box, cand_cls, out);

    (void)in_sizes; (void)n_in; (void)ws_size;
}